// PPISPSlangtorch_90752658964747
// MI455X (gfx1250) — compile-verified
//
#include <hip/hip_runtime.h>
#include <math.h>

typedef float  f4 __attribute__((ext_vector_type(4)));
typedef int    i4 __attribute__((ext_vector_type(4)));

#define FT_STRIDE 12   // per-frame fused matrix M'' = s * M * diag(gr,1,gb):
                       // [0..3]=M00,M01,M02,M10 | [4..7]=M11,M12,M20,M21 | [8]=M22, pad..
#define CT_STRIDE 12   // per-(cam,ch): cx, cy, a1, a2 | A, B, C, D | a3, pad..
#define NCAMCH    24   // 8 cameras * 3 channels
#define PPISP_EPS 1e-6f

// ---------------------------------------------------------------------------
// Kernel 1: tiny precompute. Fuses exposure scale (2^EV), WB gains (exp) and
// the off-diagonal mixing matrix into one 3x3 per frame, and softplus's the
// CRF params per (camera, channel). Removes ALL exp/softplus and 5 muls/pixel
// from the hot path.
// ---------------------------------------------------------------------------
__global__ void ppisp_precompute(const float* __restrict__ expo,
                                 const float* __restrict__ vig,
                                 const float* __restrict__ colp,
                                 const float* __restrict__ crf,
                                 float* __restrict__ frameTab,
                                 float* __restrict__ camTab,
                                 int nframes, int ncamch)
{
    int i = blockIdx.x * blockDim.x + threadIdx.x;
    if (i < nframes) {
        const float* cp = colp + i * 8;
        float s  = exp2f(expo[i]);     // exposure scale 2^EV
        float gr = s * expf(cp[0]);    // s * WB gain r   (column-0 scale)
        float gb = s * expf(cp[1]);    // s * WB gain b   (column-2 scale)
        float* ft = frameTab + i * FT_STRIDE;
        // M''[i][j] = s * M[i][j] * g[j],  M = [[1,c2,c3],[c4,1,c5],[c6,c7,1]]
        ft[0]  = gr;            // M00
        ft[1]  = s  * cp[2];    // M01
        ft[2]  = gb * cp[3];    // M02
        ft[3]  = gr * cp[4];    // M10
        ft[4]  = s;             // M11
        ft[5]  = gb * cp[5];    // M12
        ft[6]  = gr * cp[6];    // M20
        ft[7]  = s  * cp[7];    // M21
        ft[8]  = gb;            // M22
        ft[9]  = 0.f; ft[10] = 0.f; ft[11] = 0.f;
    }
    if (i < ncamch) {
        const float* vg = vig + i * 5;
        const float* cf = crf + i * 4;
        float* ct = camTab + i * CT_STRIDE;
        ct[0] = vg[0];                           // cx
        ct[1] = vg[1];                           // cy
        ct[2] = vg[2];                           // a1
        ct[3] = vg[3];                           // a2
        ct[4] = log1pf(expf(cf[0])) + 0.3f;      // A  (softplus+0.3)
        ct[5] = log1pf(expf(cf[1])) + 0.3f;      // B
        ct[6] = log1pf(expf(cf[2])) + 0.1f;      // C
        ct[7] = cf[3];                           // D
        ct[8] = vg[4];                           // a3
        ct[9] = 0.f; ct[10] = 0.f; ct[11] = 0.f;
    }
}

// ---------------------------------------------------------------------------
// Kernel 2: main per-pixel ISP. 4 pixels per thread; all streaming traffic is
// b128 non-temporal; per-camera table staged into LDS with CDNA5 async
// global->LDS DMA; per-frame fused matrix gathered from global (cache hot).
// ---------------------------------------------------------------------------
__global__ __launch_bounds__(256) void ppisp_main(
    const float* __restrict__ rgb,
    const float* __restrict__ coords,
    const int*   __restrict__ camIdx,
    const int*   __restrict__ frmIdx,
    const float* __restrict__ frameTab,
    const float* __restrict__ camTab,
    const int*   __restrict__ pW,
    const int*   __restrict__ pH,
    float* __restrict__ out,
    int npix)
{
    __shared__ float sCam[NCAMCH * CT_STRIDE];   // 288 floats = 72 x b128

    // --- stage camTab -> LDS via CDNA5 async load-to-LDS (ASYNCcnt path) ---
    {
        int t = threadIdx.x;
        // low 32 bits of a generic pointer to LDS == LDS byte offset (aperture map)
        unsigned ldsbase = (unsigned)(unsigned long long)(void*)&sCam[0];
        if (t < (NCAMCH * CT_STRIDE) / 4) {      // 72 lanes, one b128 each
            unsigned ldsoff = ldsbase + (unsigned)t * 16u;
            unsigned goff   = (unsigned)t * 16u;
            asm volatile("global_load_async_to_lds_b128 %0, %1, %2"
                         :
                         : "v"(ldsoff), "v"(goff), "s"(camTab)
                         : "memory");
        }
        asm volatile("s_wait_asynccnt 0" ::: "memory");
        __syncthreads();
    }

    int t = blockIdx.x * blockDim.x + threadIdx.x;
    if ((long long)t * 4 >= (long long)npix) return;

    const f4* rgb4 = (const f4*)rgb;
    const f4* crd4 = (const f4*)coords;
    const i4* cam4 = (const i4*)camIdx;
    const i4* frm4 = (const i4*)frmIdx;
    f4*       out4 = (f4*)out;

    float invW = 1.0f / (float)pW[0];
    float invH = 1.0f / (float)pH[0];

    union { f4 v[3]; float s[12]; } R, O;
    union { f4 v[2]; float s[8];  } Cd;
    R.v[0] = __builtin_nontemporal_load(&rgb4[t * 3 + 0]);
    R.v[1] = __builtin_nontemporal_load(&rgb4[t * 3 + 1]);
    R.v[2] = __builtin_nontemporal_load(&rgb4[t * 3 + 2]);
    Cd.v[0] = __builtin_nontemporal_load(&crd4[t * 2 + 0]);
    Cd.v[1] = __builtin_nontemporal_load(&crd4[t * 2 + 1]);
    i4 cm = __builtin_nontemporal_load(&cam4[t]);
    i4 fr = __builtin_nontemporal_load(&frm4[t]);

#pragma unroll
    for (int p = 0; p < 4; ++p) {
        int f = fr[p];
        int c = cm[p];

        // fused per-frame 3x3 (48KB table, hot in cache)
        const float* ft = frameTab + f * FT_STRIDE;
        f4 t0 = *(const f4*)(ft);        // M00, M01, M02, M10
        f4 t1 = *(const f4*)(ft + 4);    // M11, M12, M20, M21
        float M22 = ft[8];

        float u = fmaf(Cd.s[2 * p + 0], invW, -0.5f);
        float v = fmaf(Cd.s[2 * p + 1], invH, -0.5f);

        const float* cb = sCam + c * 3 * CT_STRIDE;

        // vignetting (exposure/gains are folded into the matrix)
        float y[3];
#pragma unroll
        for (int ch = 0; ch < 3; ++ch) {
            const float* cc = cb + ch * CT_STRIDE;
            f4 q0 = *(const f4*)cc;      // cx, cy, a1, a2 (ds_load_b128)
            float a3 = cc[8];
            float du = u - q0.x;
            float dv = v - q0.y;
            float r2 = fmaf(du, du, dv * dv);
            float fall = fmaf(r2, fmaf(r2, fmaf(r2, a3, q0.w), q0.z), 1.0f);
            y[ch] = R.s[3 * p + ch] * fall;
        }

        // fused exposure * gains * 3x3 mixing
        float z[3];
        z[0] = fmaf(t0.x, y[0], fmaf(t0.y, y[1], t0.z * y[2]));
        z[1] = fmaf(t0.w, y[0], fmaf(t1.x, y[1], t1.y * y[2]));
        z[2] = fmaf(t1.z, y[0], fmaf(t1.w, y[1], M22  * y[2]));

        // per-(camera,channel) sigmoid CRF
#pragma unroll
        for (int ch = 0; ch < 3; ++ch) {
            const float* cc = cb + ch * CT_STRIDE;
            f4 q1 = *(const f4*)(cc + 4);    // A, B, C, D (ds_load_b128)
            float xc = fminf(fmaxf(z[ch], PPISP_EPS), 1.0f);
            float xa = __builtin_amdgcn_exp2f(q1.x * __builtin_amdgcn_logf(xc));
            float tt = fmaf(q1.z, 1.0f - xc, PPISP_EPS);
            float tb = __builtin_amdgcn_exp2f(q1.y * __builtin_amdgcn_logf(tt));
            O.s[3 * p + ch] = fmaf(xa, __builtin_amdgcn_rcpf(xa + tb), q1.w);
        }
    }

    __builtin_nontemporal_store(O.v[0], &out4[t * 3 + 0]);
    __builtin_nontemporal_store(O.v[1], &out4[t * 3 + 1]);
    __builtin_nontemporal_store(O.v[2], &out4[t * 3 + 2]);
}

// ---------------------------------------------------------------------------
extern "C" void kernel_launch(void* const* d_in, const int* in_sizes, int n_in,
                              void* d_out, int out_size, void* d_ws, size_t ws_size,
                              hipStream_t stream)
{
    const float* expo = (const float*)d_in[0];   // [NUM_FRAMES]
    const float* vig  = (const float*)d_in[1];   // [8,3,5]
    const float* colp = (const float*)d_in[2];   // [NUM_FRAMES,8]
    const float* crf  = (const float*)d_in[3];   // [8,3,4]
    const float* rgb  = (const float*)d_in[4];   // [N,3]
    const float* crd  = (const float*)d_in[5];   // [N,2]
    const int*   cam  = (const int*)d_in[6];     // [N]
    const int*   frm  = (const int*)d_in[7];     // [N]
    const int*   pW   = (const int*)d_in[8];     // scalar
    const int*   pH   = (const int*)d_in[9];     // scalar

    int nframes = in_sizes[0];
    int ncamch  = in_sizes[3] / 4;               // 24
    int npix    = in_sizes[4] / 3;

    float* camTab   = (float*)d_ws;                        // 24*12*4 = 1152 B (16B aligned)
    float* frameTab = (float*)((char*)d_ws + 4096);        // 1000*12*4 = 48 KB

    int preThreads = (nframes > ncamch ? nframes : ncamch);
    ppisp_precompute<<<(preThreads + 255) / 256, 256, 0, stream>>>(
        expo, vig, colp, crf, frameTab, camTab, nframes, ncamch);

    int nthreads = (npix + 3) / 4;
    ppisp_main<<<(nthreads + 255) / 256, 256, 0, stream>>>(
        rgb, crd, cam, frm, frameTab, camTab, pW, pH, (float*)d_out, npix);
}